// SinkhornDistance_37460704755787
// MI455X (gfx1250) — compile-verified
//
#include <hip/hip_runtime.h>
#include <math.h>

// ---------------------------------------------------------------------------
// Sinkhorn distance on MI455X (gfx1250).
//   x, y : (8, 1024, 64) f32      out : (8,) f32
// C built with V_WMMA_F32_16X16X4_F32 (exact f32 tensor op), Sinkhorn
// iterations as L2-resident streaming passes with hw exp, fixed 100-iteration
// schedule with a device-side convergence flag (graph-capture safe).
// ---------------------------------------------------------------------------

typedef float v2f __attribute__((ext_vector_type(2)));
typedef float v8f __attribute__((ext_vector_type(8)));

#define NB 8
#define P  1024
#define DF 64

__device__ __forceinline__ void lse_combine(float& m, float& s, float m2, float s2) {
  float nm = fmaxf(m, m2);
  s = s * __expf(m - nm) + s2 * __expf(m2 - nm);
  m = nm;
}

// ---- init: zero u, v, err, done, out ---------------------------------------
__global__ void sk_init(float* u, float* v, float* err, int* done, float* out) {
  int t = threadIdx.x;
  for (int idx = t; idx < NB * P; idx += 256) { u[idx] = 0.f; v[idx] = 0.f; }
  if (t == 0) { *err = 0.f; *done = 0; }
  if (t < NB) out[t] = 0.f;
}

// ---- squared norms of x rows and y rows ------------------------------------
__global__ void sk_norms(const float* __restrict__ x, const float* __restrict__ y,
                         float* __restrict__ nx, float* __restrict__ ny) {
  int t = blockIdx.x * blockDim.x + threadIdx.x;
  if (t >= 2 * NB * P) return;
  const float* src = (t < NB * P) ? x : y;
  float* dst       = (t < NB * P) ? nx : ny;
  int r = (t < NB * P) ? t : t - NB * P;
  const float* p = src + (size_t)r * DF;
  float s = 0.f;
  #pragma unroll
  for (int k = 0; k < DF; ++k) { float a = p[k]; s = fmaf(a, a, s); }
  dst[r] = s;
}

// ---- cost matrix: C = (|x|^2 + |y|^2 - 2 x.y)/64 via f32 WMMA --------------
// One wave per 16x16 tile; A = X tile (16x4 f32 frag), B = Y^T tile (4x16).
// f32 16x4 A-frag layout: lanes 0-15 -> rows, VGPR{0,1} = K{0,1};
//                         lanes 16-31 -> rows, VGPR{0,1} = K{2,3}.
__global__ __launch_bounds__(256)
void sk_cost_matrix(const float* __restrict__ x, const float* __restrict__ y,
                    const float* __restrict__ nx, const float* __restrict__ ny,
                    float* __restrict__ C) {
  int lane = threadIdx.x & 31;
  int wave = threadIdx.x >> 5;
  int tile = blockIdx.x * 8 + wave;          // 8 batches * 64*64 tiles = 32768
  int b  = tile >> 12;
  int t  = tile & 4095;
  int tm = (t >> 6) << 4;
  int tn = (t & 63) << 4;
  int half = lane >> 4;
  int r    = lane & 15;

  const float* xrow = x + ((size_t)b * P + tm + r) * DF;
  const float* yrow = y + ((size_t)b * P + tn + r) * DF;

  v8f acc = {0.f, 0.f, 0.f, 0.f, 0.f, 0.f, 0.f, 0.f};
  #pragma unroll
  for (int k0 = 0; k0 < DF; k0 += 4) {
    v2f a  = *(const v2f*)(xrow + k0 + 2 * half);   // X[m][k0+2h .. +1]
    v2f bb = *(const v2f*)(yrow + k0 + 2 * half);   // Y[n][k0+2h .. +1] == B^T
    acc = __builtin_amdgcn_wmma_f32_16x16x4_f32(
        false, a, false, bb, (short)0, acc, false, false);
  }

  int col   = lane & 15;
  float nyv = ny[(size_t)b * P + tn + col];
  #pragma unroll
  for (int rr = 0; rr < 8; ++rr) {
    int m = tm + rr + 8 * half;                     // C/D frag: vgpr rr -> M
    float nxv = nx[(size_t)b * P + m];
    C[((size_t)b * P + m) * P + tn + col] =
        (nxv + nyv - 2.f * acc[rr]) * (1.f / 64.f);
  }
}

// ---- row pass: u_i = eps*(log_mu - lse_j((v_j - C_ij)/eps)) ----------------
__global__ __launch_bounds__(256)
void sk_row_update(const float* __restrict__ C, float* __restrict__ u,
                   const float* __restrict__ v, float* __restrict__ err,
                   const int* __restrict__ done) {
  if (*done) return;
  int lane = threadIdx.x & 31;
  int w = blockIdx.x * 8 + (threadIdx.x >> 5);      // 8192 waves
  int b = w >> 10, i = w & 1023;
  const float* Crow = C + ((size_t)b * P + i) * P;
  const float* vb   = v + (size_t)b * P;

  float m = -INFINITY, s = 0.f;
  for (int j = lane; j < P; j += 32) {
    float val = (vb[j] - Crow[j]) * 10.0f;          // 1/eps = 10
    lse_combine(m, s, val, 1.f);
  }
  #pragma unroll
  for (int off = 16; off; off >>= 1) {
    float m2 = __shfl_xor(m, off, 32);
    float s2 = __shfl_xor(s, off, 32);
    lse_combine(m, s, m2, s2);
  }
  if (lane == 0) {
    float lse    = m + __logf(s);
    float log_mu = __logf(1.0f / 1024.0f + 1e-8f);
    float u_new  = 0.1f * (log_mu - lse);
    float du     = fabsf(u_new - u[(size_t)b * P + i]);
    u[(size_t)b * P + i] = u_new;
    atomicAdd(err, du);
  }
}

// ---- column pass, stage 1: partial (m,s) over 256-row chunks ---------------
__global__ __launch_bounds__(256)
void sk_col_partial(const float* __restrict__ C, const float* __restrict__ u,
                    float2* __restrict__ part, const int* __restrict__ done) {
  if (*done) return;
  int blk = blockIdx.x;                 // b*16 + rc*4 + cb  (128 blocks)
  int b  = blk >> 4;
  int rc = (blk >> 2) & 3;
  int cb = blk & 3;
  int j  = cb * 256 + threadIdx.x;
  const float* ub = u + (size_t)b * P;
  const float* Cb = C + (size_t)b * P * P;

  float m = -INFINITY, s = 0.f;
  for (int i = rc * 256; i < rc * 256 + 256; ++i) {
    float val = (ub[i] - Cb[(size_t)i * P + j]) * 10.0f;
    lse_combine(m, s, val, 1.f);
  }
  part[((size_t)b * 4 + rc) * P + j] = make_float2(m, s);
}

// ---- column pass, stage 2: combine partials, write v -----------------------
__global__ void sk_col_combine(const float2* __restrict__ part,
                               float* __restrict__ v, const int* __restrict__ done) {
  if (*done) return;
  int t = blockIdx.x * blockDim.x + threadIdx.x;    // 8192 threads
  int b = t >> 10, j = t & 1023;
  float m = -INFINITY, s = 0.f;
  #pragma unroll
  for (int rc = 0; rc < 4; ++rc) {
    float2 p = part[((size_t)b * 4 + rc) * P + j];
    lse_combine(m, s, p.x, p.y);
  }
  float lse    = m + __logf(s);
  float log_nu = __logf(1.0f / 1024.0f + 1e-8f);
  v[t] = 0.1f * (log_nu - lse);
}

// ---- convergence check (matches scan's done-flag semantics) ----------------
__global__ void sk_finalize(float* err, int* done) {
  if (!*done && (*err * 0.125f) < 0.1f) *done = 1;  // mean over 8 batches
  *err = 0.f;
}

// ---- final cost: sum_ij exp((u_i+v_j-C)/eps) * C ---------------------------
__global__ __launch_bounds__(256)
void sk_cost(const float* __restrict__ C, const float* __restrict__ u,
             const float* __restrict__ v, float* __restrict__ out) {
  int lane = threadIdx.x & 31;
  int w = blockIdx.x * 8 + (threadIdx.x >> 5);
  int b = w >> 10, i = w & 1023;
  const float* Crow = C + ((size_t)b * P + i) * P;
  const float* vb   = v + (size_t)b * P;
  float ui = u[(size_t)b * P + i];

  float acc = 0.f;
  for (int j = lane; j < P; j += 32) {
    float c = Crow[j];
    acc += __expf((ui + vb[j] - c) * 10.0f) * c;
  }
  #pragma unroll
  for (int off = 16; off; off >>= 1) acc += __shfl_xor(acc, off, 32);
  if (lane == 0) atomicAdd(&out[b], acc);
}

// ---------------------------------------------------------------------------
extern "C" void kernel_launch(void* const* d_in, const int* in_sizes, int n_in,
                              void* d_out, int out_size, void* d_ws, size_t ws_size,
                              hipStream_t stream) {
  const float* x = (const float*)d_in[0];
  const float* y = (const float*)d_in[1];
  float* out = (float*)d_out;

  // workspace carve-out (~34 MB): C stays L2-resident (192 MB L2)
  float*  ws   = (float*)d_ws;
  float*  C    = ws;                                   // 8*1024*1024
  float*  u    = C  + (size_t)NB * P * P;              // 8192
  float*  v    = u  + NB * P;                          // 8192
  float*  nx   = v  + NB * P;                          // 8192
  float*  ny   = nx + NB * P;                          // 8192
  float2* part = (float2*)(ny + NB * P);               // 8*4*1024 float2
  float*  err  = (float*)(part + (size_t)NB * 4 * P);  // 1
  int*    done = (int*)(err + 1);                      // 1

  sk_init<<<1, 256, 0, stream>>>(u, v, err, done, out);
  sk_norms<<<(2 * NB * P + 255) / 256, 256, 0, stream>>>(x, y, nx, ny);
  sk_cost_matrix<<<NB * 64 * 64 / 8, 256, 0, stream>>>(x, y, nx, ny, C);

  for (int it = 0; it < 100; ++it) {
    sk_row_update<<<NB * P / 8, 256, 0, stream>>>(C, u, v, err, done);
    sk_col_partial<<<NB * 4 * 4, 256, 0, stream>>>(C, u, part, done);
    sk_col_combine<<<NB * P / 256, 256, 0, stream>>>(part, v, done);
    sk_finalize<<<1, 1, 0, stream>>>(err, done);
  }

  sk_cost<<<NB * P / 8, 256, 0, stream>>>(C, u, v, out);
}